// BP_82995948028182
// MI455X (gfx1250) — compile-verified
//
#include <hip/hip_runtime.h>
#include <math.h>

// ---- geometry constants (match the reference) ----
#define V_VIEWS 720
#define U_DET   736
#define NXY     512

constexpr double DU_D  = 1.2858;
constexpr float  DU_F  = 1.2858f;
constexpr double DSO_D = 595.0;
constexpr double DSD_D = 595.0 + 490.6;   // 1085.6
constexpr float  DSO_F = 595.0f;
constexpr float  DSD_F = 1085.6f;

typedef __attribute__((ext_vector_type(2))) float v2f;
typedef __attribute__((ext_vector_type(8))) float v8f;

// --------------------------------------------------------------------------
// Build the filtering matrix H[k][j] = c[k] * DU * hRamLak(j - k).
// The per-input-detector cosine weight c[k] = DSD/sqrt(DSD^2+us[k]^2) depends
// only on the contraction index k, so it is folded into H; the GEMM can then
// consume the raw sinogram as the A operand (one fewer pass over the data).
// --------------------------------------------------------------------------
__global__ void build_filter_matrix(float* __restrict__ H) {
  int idx = blockIdx.x * blockDim.x + threadIdx.x;
  if (idx >= U_DET * U_DET) return;
  int k = idx / U_DET;   // input detector index (contraction)
  int j = idx % U_DET;   // output detector index
  int d = j - k;         // spatial kernel offset, |d| <= U-1

  double hv = 0.0;
  if (d == 0) {
    hv = 1.0 / (4.0 * DU_D * DU_D);
  } else if ((d & 1) != 0) {           // odd offset (works for negative d too)
    double pnd = M_PI * (double)d * DU_D;
    hv = -1.0 / (pnd * pnd);
  }
  double us = ((double)k - (U_DET - 1) * 0.5) * DU_D;
  double c  = DSD_D / sqrt(DSD_D * DSD_D + us * us);  // cosine weight
  H[idx] = (float)(c * DU_D * hv);
}

// --------------------------------------------------------------------------
// Per-view (cos(beta), sin(beta)) table.
// --------------------------------------------------------------------------
__global__ void build_trig(float2* __restrict__ T) {
  int v = blockIdx.x * blockDim.x + threadIdx.x;
  if (v >= V_VIEWS) return;
  double beta = (double)v * (2.0 * M_PI / (double)V_VIEWS);
  T[v] = make_float2((float)cos(beta), (float)sin(beta));
}

// --------------------------------------------------------------------------
// Ramp filtering as a dense fp32 GEMM on the WMMA pipe:
//   Q[720,736] = S[720,736] x H[736,736]
// One wave per 16x16 output tile; K-loop of 184 V_WMMA_F32_16X16X4_F32 ops.
// Fragment layouts per CDNA5 ISA (32-bit A 16x4, B 4x16, D 16x16).
// --------------------------------------------------------------------------
__global__ void __launch_bounds__(32)
fbp_filter_gemm(const float* __restrict__ S, const float* __restrict__ H,
                float* __restrict__ Q) {
  const int tn   = blockIdx.x;        // N tile: 0..45
  const int tm   = blockIdx.y;        // M tile: 0..44
  const int lane = threadIdx.x;       // wave32
  const int m    = lane & 15;
  const int hi   = lane >> 4;         // lane half selects K/M sub-block

  // A: lane holds row (tm*16+m), K = vgpr + 2*hi  -> contiguous float2
  const float* arow = S + (tm * 16 + m) * U_DET + 2 * hi;
  // B: lane holds column (tn*16+m), rows k + (vgpr + 2*hi)
  const float* bcol = H + tn * 16 + m;

  v8f acc = {};
  for (int k = 0; k < U_DET; k += 4) {
    float2 a2 = *reinterpret_cast<const float2*>(arow + k);   // 8B aligned
    v2f a = { a2.x, a2.y };
    const float* bp = bcol + (size_t)(k + 2 * hi) * U_DET;
    v2f b = { bp[0], bp[U_DET] };
    // (neg_a, A, neg_b, B, c_mod, C, reuse_a, reuse_b)
    acc = __builtin_amdgcn_wmma_f32_16x16x4_f32(
        false, a, false, b, (short)0, acc, false, false);
  }

  // D: vgpr i -> row (tm*16 + hi*8 + i), col (tn*16 + m)
  float* qout = Q + (size_t)(tm * 16 + hi * 8) * U_DET + tn * 16 + m;
#pragma unroll
  for (int i = 0; i < 8; ++i) qout[(size_t)i * U_DET] = acc[i];
}

// --------------------------------------------------------------------------
// Weighted fan-beam backprojection. One thread per pixel, loop over views.
// Q (2.1 MB) stays L2-resident; the trig table is staged in LDS once per
// block so the inner loop is pure FMA + two gathers.
// --------------------------------------------------------------------------
__global__ void __launch_bounds__(256)
fbp_backproject(const float* __restrict__ Q, const float2* __restrict__ T,
                float* __restrict__ out) {
  __shared__ float2 trig[V_VIEWS];
  const int tid = threadIdx.y * 16 + threadIdx.x;
  for (int i = tid; i < V_VIEWS; i += 256) trig[i] = T[i];
  __syncthreads();

  const int iy = blockIdx.x * 16 + threadIdx.x;   // fast output dim (coalesced)
  const int ix = blockIdx.y * 16 + threadIdx.y;

  const float dx = 400.0f / 512.0f;               // sVoxel / nVoxel
  const float X = ((float)ix - 255.5f) * dx;      // xs[ix]  (meshgrid 'ij')
  const float Y = ((float)iy - 255.5f) * dx;      // xs[iy]

  float acc = 0.0f;
  for (int v = 0; v < V_VIEWS; ++v) {
    float2 cs = trig[v];
    float t =  X * cs.x + Y * cs.y;      // along detector
    float s = -X * cs.y + Y * cs.x;      // toward source
    float D  = DSO_F - s;
    float rD = 1.0f / D;
    float u   = DSD_F * t * rD;
    float idx = u / DU_F + (float)(U_DET - 1) * 0.5f;

    float fl  = floorf(idx);
    float i0f = fminf(fmaxf(fl, 0.0f), (float)(U_DET - 2));
    int   i0  = (int)i0f;
    float f   = idx - i0f;

    const float* qrow = Q + (size_t)v * U_DET;
    float q0 = qrow[i0];
    float q1 = qrow[i0 + 1];
    float val = fmaf(f, q1 - q0, q0);    // q0*(1-f) + q1*f

    bool inb = (idx >= 0.0f) && (idx <= (float)(U_DET - 1));
    val = inb ? val : 0.0f;

    float w = DSO_F * rD;                // DSO / D
    acc = fmaf(w * w, val, acc);
  }

  // 0.5 * dbeta scaling (full 2*pi fan scan covers each ray twice)
  out[(size_t)ix * NXY + iy] = acc * (float)(M_PI / (double)V_VIEWS);
}

// --------------------------------------------------------------------------
extern "C" void kernel_launch(void* const* d_in, const int* in_sizes, int n_in,
                              void* d_out, int out_size, void* d_ws, size_t ws_size,
                              hipStream_t stream) {
  (void)in_sizes; (void)n_in; (void)out_size; (void)ws_size;

  const float* sino = (const float*)d_in[0];   // (1,1,720,736) f32
  float* out = (float*)d_out;                  // (1,1,512,512) f32

  char* ws = (char*)d_ws;                      // ~4.3 MB used
  float*  H = (float*)ws;                                          // 736*736 f32
  float*  Q = (float*)(ws + (size_t)U_DET * U_DET * sizeof(float));// 720*736 f32
  float2* T = (float2*)(ws + ((size_t)U_DET * U_DET +
                              (size_t)V_VIEWS * U_DET) * sizeof(float)); // 720 f2

  build_filter_matrix<<<(U_DET * U_DET + 255) / 256, 256, 0, stream>>>(H);
  build_trig<<<(V_VIEWS + 255) / 256, 256, 0, stream>>>(T);

  dim3 ggemm(U_DET / 16, V_VIEWS / 16);        // (46, 45) tiles, 1 wave each
  fbp_filter_gemm<<<ggemm, 32, 0, stream>>>(sino, H, Q);

  fbp_backproject<<<dim3(NXY / 16, NXY / 16), dim3(16, 16), 0, stream>>>(Q, T, out);
}